// NNDecoder_15264313770421
// MI455X (gfx1250) — compile-verified
//
#include <hip/hip_runtime.h>
#include <hip/hip_bf16.h>
#include <math.h>

typedef __attribute__((ext_vector_type(2))) float v2f;
typedef __attribute__((ext_vector_type(8))) float v8f;

#define IN_DIM 128
#define L1_DIM 16
#define N_ETYPE 200

// -----------------------------------------------------------------------------
// Phase 1: H1 = relu(Z @ W1), H2 = relu(Z @ W2)   via V_WMMA_F32_16X16X4_F32
// One wave per 16-row tile of Z. D/C tile is 16x16 == 16 rows x L1_DIM cols.
// A (16x4 f32) layout: lanes 0-15 -> M=lane, lanes 16-31 -> M=lane-16;
//   VGPR0 holds K=0 (low half) / K=2 (high half), VGPR1 holds K=1 / K=3.
// B (4x16 f32) layout: VGPR0 lanes0-15 = row K=0 (N=0..15), lanes16-31 = K=2;
//   VGPR1 = K=1 / K=3.
// C/D: VGPR r, lanes 0-15 -> M=r, lanes 16-31 -> M=r+8; N = lane&15.
// -----------------------------------------------------------------------------
__global__ void node_proj_wmma(const float* __restrict__ z,
                               const float* __restrict__ w1,
                               const float* __restrict__ w2,
                               float* __restrict__ h1,
                               float* __restrict__ h2,
                               int n_tiles)
{
    const int lane = threadIdx.x & 31;
    const int wave = threadIdx.x >> 5;
    const int tile = blockIdx.x * (blockDim.x >> 5) + wave;
    if (tile >= n_tiles) return;              // uniform per wave: EXEC all-1s inside

    const int m     = lane & 15;              // A row / D row-base selector
    const int khalf = (lane >> 4) << 1;       // 0 for lanes 0-15, 2 for lanes 16-31
    const int n     = lane & 15;              // B column
    const float* __restrict__ zrow = z + (size_t)(tile * 16 + m) * IN_DIM;

    v8f c1 = {};
    v8f c2 = {};

    #pragma unroll 4
    for (int kk = 0; kk < IN_DIM; kk += 4) {
        // A fragment: two consecutive K values for this lane (8B-aligned pair)
        v2f a;
        a.x = zrow[kk + khalf];
        a.y = zrow[kk + khalf + 1];
        // B fragments for W1 and W2 (row-major [128,16], stride 16 floats)
        v2f b1, b2;
        b1.x = w1[(kk + khalf)     * L1_DIM + n];
        b1.y = w1[(kk + khalf + 1) * L1_DIM + n];
        b2.x = w2[(kk + khalf)     * L1_DIM + n];
        b2.y = w2[(kk + khalf + 1) * L1_DIM + n];
        // D = A*B + C ; same A feeds both accumulators
        c1 = __builtin_amdgcn_wmma_f32_16x16x4_f32(false, a, false, b1,
                                                   (short)0, c1, false, false);
        c2 = __builtin_amdgcn_wmma_f32_16x16x4_f32(false, a, false, b2,
                                                   (short)0, c2, false, false);
    }

    // ReLU + store D tile: row = r + 8*(lane>>4), col = lane&15
    const int rbase = tile * 16 + ((lane >> 4) << 3);
    #pragma unroll
    for (int r = 0; r < 8; ++r) {
        float v1 = c1[r]; v1 = v1 > 0.0f ? v1 : 0.0f;
        float v2 = c2[r]; v2 = v2 > 0.0f ? v2 : 0.0f;
        const size_t off = (size_t)(rbase + r) * L1_DIM + n;
        h1[off] = v1;
        h2[off] = v2;
    }
}

// Scalar tail for node counts not divisible by 16 (not hit for N=50000).
__global__ void node_proj_tail(const float* __restrict__ z,
                               const float* __restrict__ w1,
                               const float* __restrict__ w2,
                               float* __restrict__ h1,
                               float* __restrict__ h2,
                               int start_row, int n_nodes)
{
    int idx = blockIdx.x * blockDim.x + threadIdx.x;   // (row-start)*16 + col
    int row = start_row + (idx >> 4);
    int col = idx & 15;
    if (row >= n_nodes) return;
    const float* zr = z + (size_t)row * IN_DIM;
    float a1 = 0.0f, a2 = 0.0f;
    for (int k = 0; k < IN_DIM; ++k) {
        float zv = zr[k];
        a1 += zv * w1[k * L1_DIM + col];
        a2 += zv * w2[k * L1_DIM + col];
    }
    h1[(size_t)row * L1_DIM + col] = a1 > 0.0f ? a1 : 0.0f;
    h2[(size_t)row * L1_DIM + col] = a2 > 0.0f ? a2 : 0.0f;
}

// -----------------------------------------------------------------------------
// Phase 2: per-edge gather + 16-wide dot + sigmoid. Memory bound; H1/H2 (6.4MB)
// live in L2 (192MB), edge-type tables staged in LDS (25.6KB / block).
// -----------------------------------------------------------------------------
__global__ void edge_score(const float* __restrict__ h1,
                           const float* __restrict__ h2,
                           const float* __restrict__ w1_l2,
                           const float* __restrict__ w2_l2,
                           const int* __restrict__ e_src,
                           const int* __restrict__ e_dst,
                           const int* __restrict__ e_type,
                           float* __restrict__ out,
                           int n_edges)
{
    __shared__ float lw1[N_ETYPE * L1_DIM];
    __shared__ float lw2[N_ETYPE * L1_DIM];
    for (int i = threadIdx.x; i < N_ETYPE * L1_DIM; i += blockDim.x) {
        lw1[i] = w1_l2[i];
        lw2[i] = w2_l2[i];
    }
    __syncthreads();

    const int e = blockIdx.x * blockDim.x + threadIdx.x;
    if (e >= n_edges) return;

    const int s = e_src[e];
    const int d = e_dst[e];
    const int t = e_type[e];

    const float4* __restrict__ hs = (const float4*)(h1 + (size_t)s * L1_DIM);
    const float4* __restrict__ hd = (const float4*)(h2 + (size_t)d * L1_DIM);
    const float4* __restrict__ ws = (const float4*)(lw1 + t * L1_DIM);
    const float4* __restrict__ wd = (const float4*)(lw2 + t * L1_DIM);

    float acc = 0.0f;
    #pragma unroll
    for (int i = 0; i < L1_DIM / 4; ++i) {
        float4 a = hs[i], b = ws[i];
        acc += a.x * b.x + a.y * b.y + a.z * b.z + a.w * b.w;
        float4 c = hd[i], w = wd[i];
        acc += c.x * w.x + c.y * w.y + c.z * w.z + c.w * w.w;
    }
    out[e] = 1.0f / (1.0f + __expf(-acc));
}

// -----------------------------------------------------------------------------
// Inputs (setup_inputs order): z, w1_l1, w1_l2, w2_l1, w2_l2, edge_index, edge_type
// -----------------------------------------------------------------------------
extern "C" void kernel_launch(void* const* d_in, const int* in_sizes, int n_in,
                              void* d_out, int out_size, void* d_ws, size_t ws_size,
                              hipStream_t stream)
{
    const float* z     = (const float*)d_in[0];
    const float* w1_l1 = (const float*)d_in[1];
    const float* w1_l2 = (const float*)d_in[2];
    const float* w2_l1 = (const float*)d_in[3];
    const float* w2_l2 = (const float*)d_in[4];
    const int*   eidx  = (const int*)d_in[5];   // [2, E]: row0=src, row1=dst
    const int*   etyp  = (const int*)d_in[6];   // [E]
    float*       out   = (float*)d_out;

    const int n_nodes = in_sizes[0] / IN_DIM;
    const int n_edges = in_sizes[6];

    float* h1 = (float*)d_ws;                         // [n_nodes, 16]
    float* h2 = h1 + (size_t)n_nodes * L1_DIM;        // [n_nodes, 16]

    // Phase 1: WMMA node projection (8 waves / block, 1 tile of 16 rows / wave)
    const int n_tiles = n_nodes / 16;
    if (n_tiles > 0) {
        const int waves_per_block = 8;
        const int blocks = (n_tiles + waves_per_block - 1) / waves_per_block;
        node_proj_wmma<<<blocks, waves_per_block * 32, 0, stream>>>(
            z, w1_l1, w2_l1, h1, h2, n_tiles);
    }
    const int tail_rows = n_nodes - n_tiles * 16;
    if (tail_rows > 0) {
        const int tail_threads = tail_rows * L1_DIM;
        node_proj_tail<<<(tail_threads + 127) / 128, 128, 0, stream>>>(
            z, w1_l1, w2_l1, h1, h2, n_tiles * 16, n_nodes);
    }

    // Phase 2: edge scoring
    const int* e_src = eidx;
    const int* e_dst = eidx + n_edges;
    edge_score<<<(n_edges + 255) / 256, 256, 0, stream>>>(
        h1, h2, w1_l2, w2_l2, e_src, e_dst, etyp, out, n_edges);
}